// SimpleTensorAggregateLayer_6734508720705
// MI455X (gfx1250) — compile-verified
//
#include <hip/hip_runtime.h>

typedef __attribute__((ext_vector_type(2))) float v2f;
typedef __attribute__((ext_vector_type(8))) float v8f;

#define NCH 16
#define NRBF 16
#define R_CUTF 5.0f
#define PI_F 3.14159265358979323846f

__global__ __launch_bounds__(256) void zero_out_kernel(float* __restrict__ out, int n) {
    int i = blockIdx.x * blockDim.x + threadIdx.x;
    if (i < n) out[i] = 0.0f;
}

// One wave32 handles a tile of 16 edges x 16 channels.
// fn = (rbf @ W + b) * 0.1 computed with V_WMMA_F32_16X16X4_F32 (exact f32).
__global__ __launch_bounds__(256) void edge_msg_kernel(
    const float* __restrict__ pos,
    const float* __restrict__ x0,
    const float* __restrict__ x1,
    const float* __restrict__ x2,
    const float* __restrict__ W0, const float* __restrict__ b0,
    const float* __restrict__ W1, const float* __restrict__ b1,
    const float* __restrict__ W2, const float* __restrict__ b2,
    const int*   __restrict__ idx_i,
    const int*   __restrict__ idx_j,
    float* __restrict__ out0,
    float* __restrict__ out1,
    float* __restrict__ out2,
    int n_edges)
{
    const int lane = threadIdx.x & 31;
    const int wave = (blockIdx.x * blockDim.x + threadIdx.x) >> 5;
    const int half = lane >> 4;        // 0: lanes 0-15, 1: lanes 16-31
    const int m16  = lane & 15;
    const int ebase = wave << 4;       // 16 edges per wave

    // ---- per-lane edge geometry (both halves compute edge m16 redundantly) ----
    int e = ebase + m16;
    if (e >= n_edges) e = n_edges - 1;     // clamp (loads only; atomics guarded)
    const int ii = idx_i[e];
    const int jj = idx_j[e];
    const float rx = pos[jj*3+0] - pos[ii*3+0];
    const float ry = pos[jj*3+1] - pos[ii*3+1];
    const float rz = pos[jj*3+2] - pos[ii*3+2];
    const float d  = sqrtf(rx*rx + ry*ry + rz*rz + 1e-12f);
    const float inv_d = 1.0f / d;
    const float ux = rx*inv_d, uy = ry*inv_d, uz = rz*inv_d;
    const float dc = fminf(d, R_CUTF);
    const float fcut = 0.5f * (__cosf(PI_F * dc / R_CUTF) + 1.0f);

    // radial phase step and common amplitude for this lane's edge
    const float phase = (PI_F / R_CUTF) * d;   // rbf[k] = sin((k+1)*phase) * amp
    const float amp   = inv_d * fcut;

    // ---- B operands: W chunks in f32 16x16x4 B-layout ----
    // lanes 0-15 hold rows K=4kk+0 (.x), 4kk+1 (.y); lanes 16-31 rows 4kk+2, 4kk+3
    v2f BW0[4], BW1[4], BW2[4];
#pragma unroll
    for (int kk = 0; kk < 4; ++kk) {
        const int kr = 4*kk + 2*half;
        BW0[kk].x = W0[(kr+0)*NCH + m16]; BW0[kk].y = W0[(kr+1)*NCH + m16];
        BW1[kk].x = W1[(kr+0)*NCH + m16]; BW1[kk].y = W1[(kr+1)*NCH + m16];
        BW2[kk].x = W2[(kr+0)*NCH + m16]; BW2[kk].y = W2[(kr+1)*NCH + m16];
    }

    // ---- A operands: compute ONLY the 8 rbf values this lane feeds ----
    // chunk kk needs k = 4*kk + 2*half + {0,1}  ->  n = k+1 as a float, no
    // dynamically-indexed private array (avoids cndmask select trees).
    const float nbase = (float)(2*half + 1);
    v2f Arb[4];
#pragma unroll
    for (int kk = 0; kk < 4; ++kk) {
        const float n0 = nbase + (float)(4*kk);
        Arb[kk].x = __sinf(n0          * phase) * amp;
        Arb[kk].y = __sinf((n0 + 1.0f) * phase) * amp;
    }

    // ---- accumulators seeded with bias (D/C layout: lane -> channel m16) ----
    const float bb0 = b0[m16], bb1 = b1[m16], bb2 = b2[m16];
    v8f f0v = {bb0, bb0, bb0, bb0, bb0, bb0, bb0, bb0};
    v8f f1v = {bb1, bb1, bb1, bb1, bb1, bb1, bb1, bb1};
    v8f f2v = {bb2, bb2, bb2, bb2, bb2, bb2, bb2, bb2};

    // ---- fn = rbf @ W + b via 4 chained K=4 f32 WMMAs per weight ----
#pragma unroll
    for (int kk = 0; kk < 4; ++kk) {
        f0v = __builtin_amdgcn_wmma_f32_16x16x4_f32(false, Arb[kk], false, BW0[kk], (short)0, f0v, false, false);
        f1v = __builtin_amdgcn_wmma_f32_16x16x4_f32(false, Arb[kk], false, BW1[kk], (short)0, f1v, false, false);
        f2v = __builtin_amdgcn_wmma_f32_16x16x4_f32(false, Arb[kk], false, BW2[kk], (short)0, f2v, false, false);
    }
    // fold segment-sum normalization (1/10) into fn (everything is linear in fn)
#pragma unroll
    for (int v = 0; v < 8; ++v) {
        f0v[v] *= 0.1f; f1v[v] *= 0.1f; f2v[v] *= 0.1f;
    }

    // ---- aggregation: lane owns channel c = m16, edges v + 8*half (D layout) ----
    const int c = m16;
#pragma unroll
    for (int v = 0; v < 8; ++v) {
        const int m = v + 8*half;               // edge slot within tile
        const float f0 = f0v[v], f1 = f1v[v], f2 = f2v[v];
        const float eux = __shfl(ux, m, 32);
        const float euy = __shfl(uy, m, 32);
        const float euz = __shfl(uz, m, 32);
        const int   ei  = __shfl(ii, m, 32);
        const int   ej  = __shfl(jj, m, 32);
        const int   eg  = ebase + m;

        const float* px1 = x1 + ((size_t)ej*NCH + c)*3;
        const float* px2 = x2 + ((size_t)ej*NCH + c)*9;
        const float a0 = x0[(size_t)ej*NCH + c];
        float a1[3], a2[9];
#pragma unroll
        for (int t = 0; t < 3; ++t) a1[t] = px1[t];
#pragma unroll
        for (int t = 0; t < 9; ++t) a2[t] = px2[t];

        const float u3[3] = {eux, euy, euz};
        const float s1 = a1[0]*eux + a1[1]*euy + a1[2]*euz;          // x1 . u
        float yv[3];                                                 // y_u = sum_v x2[u][v] u_v
#pragma unroll
        for (int t = 0; t < 3; ++t)
            yv[t] = a2[t*3+0]*eux + a2[t*3+1]*euy + a2[t*3+2]*euz;
        const float q = yv[0]*eux + yv[1]*euy + yv[2]*euz;           // u . x2 . u

        if (eg < n_edges) {
            // out0_c += f0*x0 + f1*(x1.u) + f2*(u.x2.u)
            atomicAdd(out0 + (size_t)ei*NCH + c, f0*a0 + f1*s1 + f2*q);

            // out1_cu += f0*x1_u + (f1*x0 + f2*(x1.u))*u_u + f1*y_u
            const float t01 = f1*a0 + f2*s1;
            float* po1 = out1 + ((size_t)ei*NCH + c)*3;
#pragma unroll
            for (int uu = 0; uu < 3; ++uu)
                atomicAdd(po1 + uu, f0*a1[uu] + t01*u3[uu] + f1*yv[uu]);

            // out2_cuv += f0*x2_uv + (f1*x1_u + f2*(x0*u_u + y_u))*u_v
            float* po2 = out2 + ((size_t)ei*NCH + c)*9;
#pragma unroll
            for (int uu = 0; uu < 3; ++uu) {
                const float g = f1*a1[uu] + f2*(a0*u3[uu] + yv[uu]);
#pragma unroll
                for (int vv = 0; vv < 3; ++vv)
                    atomicAdd(po2 + uu*3 + vv, f0*a2[uu*3+vv] + g*u3[vv]);
            }
        }
    }
}

extern "C" void kernel_launch(void* const* d_in, const int* in_sizes, int n_in,
                              void* d_out, int out_size, void* d_ws, size_t ws_size,
                              hipStream_t stream) {
    const float* pos = (const float*)d_in[0];
    const float* x0  = (const float*)d_in[1];
    const float* x1  = (const float*)d_in[2];
    const float* x2  = (const float*)d_in[3];
    const float* W0  = (const float*)d_in[4];
    const float* b0  = (const float*)d_in[5];
    const float* W1  = (const float*)d_in[6];
    const float* b1  = (const float*)d_in[7];
    const float* W2  = (const float*)d_in[8];
    const float* b2  = (const float*)d_in[9];
    const int* idx_i = (const int*)d_in[10];
    const int* idx_j = (const int*)d_in[11];

    const int n_edges = in_sizes[10];
    const int n_atoms = in_sizes[1] / NCH;

    float* out0 = (float*)d_out;                                   // (A,16)
    float* out1 = out0 + (size_t)n_atoms * NCH;                    // (A,16,3)
    float* out2 = out1 + (size_t)n_atoms * NCH * 3;                // (A,16,3,3)

    // zero-init outputs (harness poisons d_out)
    {
        int n = out_size;
        int blocks = (n + 255) / 256;
        zero_out_kernel<<<blocks, 256, 0, stream>>>((float*)d_out, n);
    }

    // one wave (32 threads) per 16-edge tile
    const int tiles   = (n_edges + 15) / 16;
    const int threads = tiles * 32;
    const int blocks  = (threads + 255) / 256;
    edge_msg_kernel<<<blocks, 256, 0, stream>>>(
        pos, x0, x1, x2, W0, b0, W1, b1, W2, b2, idx_i, idx_j,
        out0, out1, out2, n_edges);
}